// H2GCN_5342939316790
// MI455X (gfx1250) — compile-verified
//
#include <hip/hip_runtime.h>

// Problem sizes (fixed by the reference)
#define NN   100000
#define EE   1600000
#define INC  128
#define HID  64
#define OUTC 16
#define MT   5          // M-tiles per wave: 100000 = 1250 * 5 * 16 (exact)

typedef __attribute__((ext_vector_type(2))) float v2f;
typedef __attribute__((ext_vector_type(8))) float v8f;

// ---------------------------------------------------------------------------
// Elementwise / edge kernels
// ---------------------------------------------------------------------------
__global__ void fill_zero4(float4* __restrict__ p, int n4) {
    int i = blockIdx.x * blockDim.x + threadIdx.x;
    if (i < n4) p[i] = make_float4(0.f, 0.f, 0.f, 0.f);
}

__global__ void deg_kernel(const long long* __restrict__ col,
                           float* __restrict__ deg, int E) {
    int e = blockIdx.x * blockDim.x + threadIdx.x;
    if (e < E) atomicAdd(&deg[(int)col[e]], 1.0f);
}

__global__ void dinv_kernel(const float* __restrict__ deg,
                            float* __restrict__ dinv, int n) {
    int i = blockIdx.x * blockDim.x + threadIdx.x;
    if (i < n) {
        float d = deg[i];
        dinv[i] = (d > 0.0f) ? rsqrtf(d) : 0.0f;
    }
}

__global__ void norm_kernel(const long long* __restrict__ row,
                            const long long* __restrict__ col,
                            const float* __restrict__ dinv,
                            float* __restrict__ nrm, int E) {
    int e = blockIdx.x * blockDim.x + threadIdx.x;
    if (e < E) nrm[e] = dinv[(int)row[e]] * dinv[(int)col[e]];
}

// For each edge, gather hw[row]*norm and scatter-add into agg[col].
// 16 consecutive threads cover one edge's contiguous 256B feature row:
// coalesced b128 gather + 4 float atomics per thread (L2-resident).
__global__ void scatter_edges(const float* __restrict__ hw,
                              const long long* __restrict__ row,
                              const long long* __restrict__ col,
                              const float* __restrict__ nrm,
                              float* __restrict__ agg, int E) {
    unsigned t = blockIdx.x * blockDim.x + threadIdx.x;
    unsigned e = t >> 4;
    if (e >= (unsigned)E) return;
    int f  = (int)(t & 15u) << 2;
    int rs = (int)row[e];
    int cd = (int)col[e];
    float nm = nrm[e];
    const float4 v = *reinterpret_cast<const float4*>(hw + (size_t)rs * HID + f);
    float* dst = agg + (size_t)cd * HID + f;
    atomicAdd(dst + 0, v.x * nm);
    atomicAdd(dst + 1, v.y * nm);
    atomicAdd(dst + 2, v.z * nm);
    atomicAdd(dst + 3, v.w * nm);
}

__global__ void bias_relu4(float4* __restrict__ h,
                           const float* __restrict__ b, int n4) {
    int i = blockIdx.x * blockDim.x + threadIdx.x;
    if (i >= n4) return;
    const float4 bb = *reinterpret_cast<const float4*>(b + ((i << 2) & (HID - 1)));
    float4 v = h[i];
    v.x = fmaxf(v.x + bb.x, 0.0f);
    v.y = fmaxf(v.y + bb.y, 0.0f);
    v.z = fmaxf(v.z + bb.z, 0.0f);
    v.w = fmaxf(v.w + bb.w, 0.0f);
    h[i] = v;
}

// ---------------------------------------------------------------------------
// fp32 WMMA GEMM: out[M,NC] = act(A[M,K] @ W[K,NC] + bias)
// One wave per (MT*16) x 16 output block. Per K-step: 1 B tile (reused by all
// MT accumulators) + MT A tiles, staged into distinct locals BEFORE the MT
// WMMAs so loads overlap compute (staggered s_wait_loadcnt, not 0x0).
// K loop fully unrolled (K compile-time) -> base reg + immediate offsets,
// no loop-carried address math.
//
// V_WMMA_F32_16X16X4_F32 register layouts (ISA 7.12.2):
//   A 16x4 : lanes 0-15 hold {K=k0,k0+1}, lanes 16-31 hold {K=k0+2,k0+3},
//            row M = lane & 15
//   B 4x16 : mirrored (col N = lane & 15, same K split by lane>=16)
//   C/D    : VGPR j -> row (j + 8*(lane>>4)), col (lane & 15)
// ---------------------------------------------------------------------------
template <int K, int NC, bool RELU, bool BIAS>
__global__ void wmma_gemm_f32(const float* __restrict__ A,
                              const float* __restrict__ W,
                              const float* __restrict__ bias,
                              float* __restrict__ out) {
    const int lane  = threadIdx.x;          // 0..31
    const int r     = lane & 15;
    const int kh    = lane >> 4;            // 0 or 1
    const int mbase = blockIdx.x * (16 * MT);
    const int nbase = blockIdx.y * 16;

    // Per-lane base pointers (constant across the unrolled K loop).
    const v2f* arow[MT];
#pragma unroll
    for (int t = 0; t < MT; ++t)
        arow[t] = reinterpret_cast<const v2f*>(
            A + (size_t)(mbase + t * 16 + r) * K + 2 * kh);
    const float* wcol = W + nbase + r;

    v8f acc[MT];
#pragma unroll
    for (int t = 0; t < MT; ++t) acc[t] = (v8f){};

#pragma unroll
    for (int k0 = 0; k0 < K; k0 += 4) {
        const int ka = k0 + 2 * kh;
        v2f b;
        b.x = wcol[(size_t)ka * NC];
        b.y = wcol[(size_t)(ka + 1) * NC];
        v2f a[MT];
#pragma unroll
        for (int t = 0; t < MT; ++t) a[t] = arow[t][k0 / 2];
#pragma unroll
        for (int t = 0; t < MT; ++t)
            acc[t] = __builtin_amdgcn_wmma_f32_16x16x4_f32(
                false, a[t], false, b, (short)0, acc[t], false, false);
    }

    const int   n  = nbase + r;
    const float bb = BIAS ? bias[n] : 0.0f;
#pragma unroll
    for (int t = 0; t < MT; ++t) {
#pragma unroll
        for (int j = 0; j < 8; ++j) {
            const int m = mbase + t * 16 + j + 8 * kh;
            float v = acc[t][j] + bb;
            if (RELU) v = fmaxf(v, 0.0f);
            out[(size_t)m * NC + n] = v;
        }
    }
}

// Final layer: out[M,16] = concat(h0,h1,h2)[M,192] @ Wo[192,16] + bo.
// Concat read virtually: three uniform sub-loops, one per h buffer.
__global__ void wmma_out_gemm(const float* __restrict__ h0,
                              const float* __restrict__ h1,
                              const float* __restrict__ h2,
                              const float* __restrict__ Wo,
                              const float* __restrict__ bo,
                              float* __restrict__ out) {
    const int lane  = threadIdx.x;
    const int r     = lane & 15;
    const int kh    = lane >> 4;
    const int mbase = blockIdx.x * (16 * MT);

    v8f acc[MT];
#pragma unroll
    for (int t = 0; t < MT; ++t) acc[t] = (v8f){};

    const float* hs[3] = {h0, h1, h2};
#pragma unroll
    for (int phase = 0; phase < 3; ++phase) {
        const float* Wp = Wo + (size_t)phase * HID * OUTC + r;
        const v2f* arow[MT];
#pragma unroll
        for (int t = 0; t < MT; ++t)
            arow[t] = reinterpret_cast<const v2f*>(
                hs[phase] + (size_t)(mbase + t * 16 + r) * HID + 2 * kh);
#pragma unroll
        for (int k0 = 0; k0 < HID; k0 += 4) {
            const int ka = k0 + 2 * kh;
            v2f b;
            b.x = Wp[(size_t)ka * OUTC];
            b.y = Wp[(size_t)(ka + 1) * OUTC];
            v2f a[MT];
#pragma unroll
            for (int t = 0; t < MT; ++t) a[t] = arow[t][k0 / 2];
#pragma unroll
            for (int t = 0; t < MT; ++t)
                acc[t] = __builtin_amdgcn_wmma_f32_16x16x4_f32(
                    false, a[t], false, b, (short)0, acc[t], false, false);
        }
    }

    const float bb = bo[r];
#pragma unroll
    for (int t = 0; t < MT; ++t) {
#pragma unroll
        for (int j = 0; j < 8; ++j) {
            const int m = mbase + t * 16 + j + 8 * kh;
            out[(size_t)m * OUTC + r] = acc[t][j] + bb;
        }
    }
}

// ---------------------------------------------------------------------------
// Host-side orchestration (graph-capture safe: only kernel launches on stream)
// ---------------------------------------------------------------------------
extern "C" void kernel_launch(void* const* d_in, const int* in_sizes, int n_in,
                              void* d_out, int out_size, void* d_ws, size_t ws_size,
                              hipStream_t stream) {
    (void)in_sizes; (void)n_in; (void)out_size; (void)ws_size;

    const float*     x   = (const float*)d_in[0];
    const long long* ei  = (const long long*)d_in[1];   // int64 edge_index [2,E]
    const float*     W0  = (const float*)d_in[2];
    const float*     b0  = (const float*)d_in[3];
    const float*     W1  = (const float*)d_in[4];
    const float*     b1  = (const float*)d_in[5];
    const float*     W2  = (const float*)d_in[6];
    const float*     b2  = (const float*)d_in[7];
    const float*     Wo  = (const float*)d_in[8];
    const float*     bo  = (const float*)d_in[9];
    float*           out = (float*)d_out;

    const long long* row = ei;              // sources
    const long long* col = ei + EE;         // targets

    // Workspace layout (floats): deg | dinv | norm | h0 | h1 | h2 | hw
    float* ws   = (float*)d_ws;
    float* deg  = ws;
    float* dinv = deg  + NN;
    float* nrm  = dinv + NN;
    float* h0   = nrm  + EE;
    float* h1   = h0   + (size_t)NN * HID;
    float* h2   = h1   + (size_t)NN * HID;
    float* hw   = h2   + (size_t)NN * HID;

    const int TB     = 256;
    const int gtiles = NN / (16 * MT);      // 1250, exact

    // Zero accumulation buffers (h1,h2 contiguous, 4-aligned) and deg.
    {
        int n4 = (2 * NN * HID) / 4;
        fill_zero4<<<(n4 + TB - 1) / TB, TB, 0, stream>>>((float4*)h1, n4);
        int d4 = NN / 4;
        fill_zero4<<<(d4 + TB - 1) / TB, TB, 0, stream>>>((float4*)deg, d4);
    }

    // Degree over targets -> dinv -> per-edge norm.
    deg_kernel<<<(EE + TB - 1) / TB, TB, 0, stream>>>(col, deg, EE);
    dinv_kernel<<<(NN + TB - 1) / TB, TB, 0, stream>>>(deg, dinv, NN);
    norm_kernel<<<(EE + TB - 1) / TB, TB, 0, stream>>>(row, col, dinv, nrm, EE);

    // h0 = relu(x @ W0 + b0)
    wmma_gemm_f32<INC, HID, true, true>
        <<<dim3(gtiles, HID / 16), 32, 0, stream>>>(x, W0, b0, h0);

    // Layer 1: hw = h0 @ W1 ; h1 = relu(scatter(norm * hw[row] -> col) + b1)
    wmma_gemm_f32<HID, HID, false, false>
        <<<dim3(gtiles, HID / 16), 32, 0, stream>>>(h0, W1, nullptr, hw);
    {
        unsigned threads = (unsigned)EE * 16u;
        scatter_edges<<<(threads + TB - 1) / TB, TB, 0, stream>>>(
            hw, row, col, nrm, h1, EE);
    }
    bias_relu4<<<((NN * HID / 4) + TB - 1) / TB, TB, 0, stream>>>(
        (float4*)h1, b1, NN * HID / 4);

    // Layer 2: hw = h1 @ W2 ; h2 = relu(scatter + b2)
    wmma_gemm_f32<HID, HID, false, false>
        <<<dim3(gtiles, HID / 16), 32, 0, stream>>>(h1, W2, nullptr, hw);
    {
        unsigned threads = (unsigned)EE * 16u;
        scatter_edges<<<(threads + TB - 1) / TB, TB, 0, stream>>>(
            hw, row, col, nrm, h2, EE);
    }
    bias_relu4<<<((NN * HID / 4) + TB - 1) / TB, TB, 0, stream>>>(
        (float4*)h2, b2, NN * HID / 4);

    // out = concat(h0,h1,h2) @ Wo + bo
    wmma_out_gemm<<<dim3(gtiles), 32, 0, stream>>>(h0, h1, h2, Wo, bo, out);
}